// AttentionModel_35785667510838
// MI455X (gfx1250) — compile-verified
//
#include <hip/hip_runtime.h>
#include <math.h>
#include <stdint.h>

typedef __attribute__((ext_vector_type(2))) float v2f;
typedef __attribute__((ext_vector_type(8))) float v8f;

namespace {

constexpr int BB = 128, NN = 512, DD = 128, HH = 8, DKK = 16, SS = 512;
constexpr float NORMF = 11.313708498984760f; // sqrt(128)
constexpr float CLIPF = 10.0f;

// D = A(16xK) * B(Kx16) + C using V_WMMA_F32_16X16X4_F32 (full f32 precision).
__device__ __forceinline__ v8f wmma4(v2f a, v2f b, v8f c) {
  return __builtin_amdgcn_wmma_f32_16x16x4_f32(
      /*neg_a=*/false, a, /*neg_b=*/false, b,
      /*c_mod=*/(short)0, c, /*reuse_a=*/false, /*reuse_b=*/false);
}

// One wave computes a 16x16 f32 tile of A(16xK, row-major lda) * B(Kx16, row-major ldb).
// K is compile-time -> fully unrolled WMMA chain.
template <int K>
__device__ __forceinline__ v8f wave_gemm_f32(const float* __restrict__ A, int lda,
                                             const float* __restrict__ Bm, int ldb) {
  const int lane = threadIdx.x & 31;
  const int mn = lane & 15;       // M for A, N for B
  const int kh = (lane >> 4) & 1; // K-half select (lanes 16-31 hold K+2/K+3)
  v8f acc = {};
#pragma unroll
  for (int k0 = 0; k0 < K; k0 += 4) {
    const int k = k0 + 2 * kh;
    v2f a, b;
    a.x = A[(size_t)mn * lda + k];
    a.y = A[(size_t)mn * lda + k + 1];
    b.x = Bm[(size_t)k * ldb + mn];
    b.y = Bm[(size_t)(k + 1) * ldb + mn];
    acc = wmma4(a, b, acc);
  }
  return acc;
}

// Stage a 16xK f32 tile (rows 16B-aligned) into LDS via GLOBAL_LOAD_ASYNC_TO_LDS_B128
// (ASYNCcnt-tracked), then drain this wave's async queue. Caller barriers afterwards.
template <int K>
__device__ __forceinline__ void async_tile_16xK(const float* __restrict__ g, int lda,
                                                float* lds, int tid, int nthr) {
  constexpr int CH = 16 * K / 4; // b128 chunks
  constexpr int PR = K / 4;      // chunks per row
  for (int c = tid; c < CH; c += nthr) {
    const int r = c / PR, col = (c % PR) * 4;
    const uint32_t lo = (uint32_t)(uintptr_t)(lds + r * K + col); // LDS byte offset
    const uint64_t ga = (uint64_t)(uintptr_t)(g + (size_t)r * lda + col);
    asm volatile("global_load_async_to_lds_b128 %0, %1, off" ::"v"(lo), "v"(ga)
                 : "memory");
  }
  asm volatile("s_wait_asynccnt 0x0" ::: "memory");
}

// fixed = (mean_n E[b]) @ W_fixed  -- tiny, one block per batch.
__global__ void k_mean_fixed(const float* __restrict__ E, const float* __restrict__ Wf,
                             float* __restrict__ fixed) {
  __shared__ float em[DD];
  const int b = blockIdx.x;
  const int d = threadIdx.x; // 128 threads
  float s = 0.f;
  const float* Eb = E + (size_t)b * NN * DD + d;
  for (int n = 0; n < NN; ++n) s += Eb[(size_t)n * DD];
  em[d] = s * (1.0f / NN);
  __syncthreads();
  float acc = 0.f;
  for (int k = 0; k < DD; ++k) acc += em[k] * Wf[k * DD + d];
  fixed[b * DD + d] = acc;
}

// firstpos[b][n] = min step index selecting n (N if never). mask(s,n) <=> firstpos < s.
__global__ void k_firstpos(const int* __restrict__ seq, int* __restrict__ fp) {
  const int b = blockIdx.x;
  const int t = threadIdx.x; // 512 threads
  fp[b * NN + t] = NN;
  __syncthreads();
  atomicMin(&fp[b * NN + seq[b * NN + t]], t);
}

// ctx[s][b][0:2D]: step 0 -> W_placeholder; else concat(E[b,sel0], E[b,seq[s-1]]).
__global__ void k_ctx(const float* __restrict__ E, const int* __restrict__ seq,
                      const float* __restrict__ Wph, float* __restrict__ ctx) {
  const int s = blockIdx.x, b = blockIdx.y, j = threadIdx.x; // 256 threads
  float v;
  if (s == 0) {
    v = Wph[j];
  } else {
    const int node = (j < DD) ? seq[b * NN] : seq[b * NN + s - 1];
    const int jj = (j < DD) ? j : (j - DD);
    v = E[((size_t)b * NN + node) * DD + jj];
  }
  ctx[((size_t)s * BB + b) * (2 * DD) + j] = v;
}

// proj = E[b] @ W_node, scattered into WMMA-operand-ready layouts:
//   gKt[b][h][dk][n], gV[b][h][n][dk], lKt[b][d][n].
// Each 16-col tile maps entirely to one destination -> wave-uniform epilogue branch.
__global__ void __launch_bounds__(128)
k_proj(const float* __restrict__ E, const float* __restrict__ Wn,
       float* __restrict__ gKt, float* __restrict__ gV, float* __restrict__ lKt) {
  __shared__ __align__(16) float at[16 * DD];
  const int b = blockIdx.x, n0 = blockIdx.y * 16;
  const int tid = threadIdx.x, w = tid >> 5, lane = tid & 31;
  const int nc = lane & 15, kh = lane >> 4;
  const int jt = blockIdx.z * 4 + w; // 24 column tiles of W_node
  async_tile_16xK<DD>(E + ((size_t)b * NN + n0) * DD, DD, at, tid, 128);
  __syncthreads();
  v8f acc = wave_gemm_f32<DD>(at, DD, Wn + jt * 16, 3 * DD);
  if (jt < 8) { // gK, head h = jt, dk = nc (transposed store)
    float* dst = gKt + (((size_t)b * HH + jt) * DKK + nc) * NN + n0;
#pragma unroll
    for (int r = 0; r < 8; ++r) dst[r + 8 * kh] = acc[r];
  } else if (jt < 16) { // gV, head h = jt-8
    float* dst = gV + (((size_t)b * HH + (jt - 8)) * NN + n0) * DKK + nc;
#pragma unroll
    for (int r = 0; r < 8; ++r) dst[(size_t)(r + 8 * kh) * DKK] = acc[r];
  } else { // lK, col (jt-16)*16 + nc (transposed store)
    float* dst = lKt + ((size_t)b * DD + (jt - 16) * 16 + nc) * NN + n0;
#pragma unroll
    for (int r = 0; r < 8; ++r) dst[r + 8 * kh] = acc[r];
  }
}

// Generic M x 128 GEMM over 16-row tiles; A tile staged once per block via async-LDS.
// Optional per-batch bias (rows are m = s*B + b).
template <int K>
__global__ void __launch_bounds__(128)
k_gemm_bias(const float* __restrict__ A, int lda,
            const float* __restrict__ Bm, int ldb,
            const float* __restrict__ bias,
            float* __restrict__ C, int ldc) {
  __shared__ __align__(16) float at[16 * K];
  const int m0 = blockIdx.x * 16;
  const int tid = threadIdx.x, w = tid >> 5, lane = tid & 31;
  const int nc = lane & 15, kh = lane >> 4;
  async_tile_16xK<K>(A + (size_t)m0 * lda, lda, at, tid, 128);
  __syncthreads();
  for (int nt = w; nt < DD / 16; nt += 4) {
    const int nb = nt * 16;
    v8f acc = wave_gemm_f32<K>(at, K, Bm + nb, ldb);
#pragma unroll
    for (int r = 0; r < 8; ++r) {
      const int m = m0 + r + 8 * kh;
      float v = acc[r];
      if (bias) v += bias[(m & (BB - 1)) * DD + nb + nc];
      C[(size_t)m * ldc + nb + nc] = v;
    }
  }
}

// Fused attention for one (b, h, 16-step tile): compat WMMA -> masked softmax in LDS
// -> attn*V WMMA (K split over 8 waves, LDS reduce) -> heads.
__global__ void __launch_bounds__(256)
k_attn(const float* __restrict__ query, const float* __restrict__ gKt,
       const float* __restrict__ gV, const int* __restrict__ fp,
       float* __restrict__ heads) {
  __shared__ float comp[16][NN];                // 32 KB
  __shared__ float part[8][16][16];             // 8 KB
  __shared__ float red[16][16];
  __shared__ float rsum[16];
  __shared__ __align__(16) float at[16 * DKK];  // staged Q tile (1 KB)
  const int sT = blockIdx.x * 16, b = blockIdx.y, h = blockIdx.z;
  const int tid = threadIdx.x;
  const int w = tid >> 5, lane = tid & 31, nc = lane & 15, kh = lane >> 4;
  const float* Kb = gKt + ((size_t)b * HH + h) * DKK * NN; // (DK x N), ldb = N
  const int* fpb = fp + b * NN;

  async_tile_16xK<DKK>(query + ((size_t)sT * BB + b) * DD + h * DKK, BB * DD, at, tid, 256);
  __syncthreads();

  // compat = NORM * Q K^T, masked
  for (int i = 0; i < 4; ++i) {
    const int nb = (w * 4 + i) * 16;
    v8f acc = wave_gemm_f32<DKK>(at, DKK, Kb + nb, NN);
#pragma unroll
    for (int r = 0; r < 8; ++r) {
      const int row = r + 8 * kh;
      const int s = sT + row;
      const int pos = nb + nc;
      float v = NORMF * acc[r];
      if (fpb[pos] < s) v = -__builtin_inff();
      comp[row][pos] = v;
    }
  }
  __syncthreads();

  // stable softmax per row (store unnormalized exp, keep row sums)
  {
    const int row = tid >> 4, l = tid & 15;
    float mx = -__builtin_inff();
    for (int c = l; c < NN; c += 16) mx = fmaxf(mx, comp[row][c]);
    red[row][l] = mx;
    __syncthreads();
    float m2 = -__builtin_inff();
#pragma unroll
    for (int i = 0; i < 16; ++i) m2 = fmaxf(m2, red[row][i]);
    __syncthreads();
    float sm = 0.f;
    for (int c = l; c < NN; c += 16) {
      const float e = __expf(comp[row][c] - m2);
      comp[row][c] = e;
      sm += e;
    }
    red[row][l] = sm;
    __syncthreads();
    float s2 = 0.f;
#pragma unroll
    for (int i = 0; i < 16; ++i) s2 += red[row][i];
    if (l == 0) rsum[row] = s2;
  }
  __syncthreads();

  // heads = attn * V : each wave takes a 64-wide K slice, partials reduced via LDS
  {
    const float* Ap = &comp[0][w * 64];
    const float* Bp = gV + (((size_t)b * HH + h) * NN + w * 64) * DKK;
    v8f acc = wave_gemm_f32<64>(Ap, NN, Bp, DKK);
#pragma unroll
    for (int r = 0; r < 8; ++r) part[w][r + 8 * kh][nc] = acc[r];
  }
  __syncthreads();
  {
    const int r = tid >> 4, c = tid & 15;
    float v = 0.f;
#pragma unroll
    for (int i = 0; i < 8; ++i) v += part[i][r][c];
    v /= rsum[r];
    heads[(((size_t)(sT + r)) * BB + b) * DD + h * DKK + c] = v;
  }
}

// Fused logits: glimpse * lK^T -> tanh*CLIP -> mask -> log_softmax -> log_p[b][s][n]
__global__ void __launch_bounds__(256)
k_logits(const float* __restrict__ glimpse, const float* __restrict__ lKt,
         const int* __restrict__ fp, float* __restrict__ out) {
  __shared__ float lg[16][NN]; // 32 KB
  __shared__ float red[16][16];
  __shared__ __align__(16) float at[16 * DD]; // staged glimpse tile (8 KB)
  const int sT = blockIdx.x * 16, b = blockIdx.y;
  const int tid = threadIdx.x;
  const int w = tid >> 5, lane = tid & 31, nc = lane & 15, kh = lane >> 4;
  const float* Bm = lKt + (size_t)b * DD * NN; // (D x N), ldb = N
  const int* fpb = fp + b * NN;

  async_tile_16xK<DD>(glimpse + ((size_t)sT * BB + b) * DD, BB * DD, at, tid, 256);
  __syncthreads();

  for (int i = 0; i < 4; ++i) {
    const int nb = (w * 4 + i) * 16;
    v8f acc = wave_gemm_f32<DD>(at, DD, Bm + nb, NN);
#pragma unroll
    for (int r = 0; r < 8; ++r) {
      const int row = r + 8 * kh;
      const int s = sT + row;
      const int pos = nb + nc;
      float v = tanhf(NORMF * acc[r]) * CLIPF;
      if (fpb[pos] < s) v = -__builtin_inff();
      lg[row][pos] = v;
    }
  }
  __syncthreads();

  const int row = tid >> 4, l = tid & 15;
  float mx = -__builtin_inff();
  for (int c = l; c < NN; c += 16) mx = fmaxf(mx, lg[row][c]);
  red[row][l] = mx;
  __syncthreads();
  float m2 = -__builtin_inff();
#pragma unroll
  for (int i = 0; i < 16; ++i) m2 = fmaxf(m2, red[row][i]);
  __syncthreads();
  float sm = 0.f;
  for (int c = l; c < NN; c += 16) sm += __expf(lg[row][c] - m2);
  red[row][l] = sm;
  __syncthreads();
  float s2 = 0.f;
#pragma unroll
  for (int i = 0; i < 16; ++i) s2 += red[row][i];
  const float lse = m2 + __logf(s2);
  float* ob = out + ((size_t)b * SS + (sT + row)) * NN;
  for (int c = l; c < NN; c += 16) ob[c] = lg[row][c] - lse;
}

__global__ void k_copyseq(const int* __restrict__ seq, int* __restrict__ dst) {
  const int i = blockIdx.x * blockDim.x + threadIdx.x;
  dst[i] = seq[i];
}

} // namespace

extern "C" void kernel_launch(void* const* d_in, const int* in_sizes, int n_in,
                              void* d_out, int out_size, void* d_ws, size_t ws_size,
                              hipStream_t stream) {
  const float* E   = (const float*)d_in[0]; // (B,N,D)
  const int*   seq = (const int*)d_in[1];   // (B,N)
  const float* Wph = (const float*)d_in[2]; // (2D,)
  const float* Wn  = (const float*)d_in[3]; // (D,3D)
  const float* Wf  = (const float*)d_in[4]; // (D,D)
  const float* Ws  = (const float*)d_in[5]; // (2D,D)
  const float* Wo  = (const float*)d_in[6]; // (D,D)
  float* out = (float*)d_out;

  // workspace layout (floats); total ~202 MB
  const size_t seg = (size_t)BB * HH * DKK * NN; // 8,388,608 (= S*B*D = B*D*N)
  float* ws      = (float*)d_ws;
  float* fixed   = ws;                     // B*D
  int*   fp      = (int*)(ws + BB * DD);   // B*N
  float* gKt     = ws + BB * DD + BB * NN; // seg
  float* gV      = gKt + seg;              // seg
  float* lKt     = gV + seg;               // seg
  float* query   = lKt + seg;              // seg
  float* ctx     = query + seg;            // 2*seg (dead after query GEMM)
  float* heads   = ctx;                    // alias: seg
  float* glimpse = ctx + seg;              // alias: seg

  k_mean_fixed<<<BB, DD, 0, stream>>>(E, Wf, fixed);
  k_firstpos<<<BB, NN, 0, stream>>>(seq, fp);
  k_ctx<<<dim3(SS, BB), 2 * DD, 0, stream>>>(E, seq, Wph, ctx);
  k_proj<<<dim3(BB, NN / 16, 6), 128, 0, stream>>>(E, Wn, gKt, gV, lKt);
  // query = ctx @ W_step + fixed
  k_gemm_bias<2 * DD><<<(SS * BB) / 16, 128, 0, stream>>>(ctx, 2 * DD, Ws, DD, fixed,
                                                          query, DD);
  k_attn<<<dim3(SS / 16, BB, HH), 256, 0, stream>>>(query, gKt, gV, fp, heads);
  // glimpse = heads @ W_out
  k_gemm_bias<DD><<<(SS * BB) / 16, 128, 0, stream>>>(heads, DD, Wo, DD, nullptr,
                                                      glimpse, DD);
  k_logits<<<dim3(SS / 16, BB), 256, 0, stream>>>(glimpse, lKt, fp, out);
  // second tuple element: eval_seq passthrough (bit-copied int32)
  k_copyseq<<<(BB * NN) / 256, 256, 0, stream>>>(seq, (int*)out + (size_t)BB * SS * NN);
}